// LinkPredictor_41695542509975
// MI455X (gfx1250) — compile-verified
//
#include <hip/hip_runtime.h>
#include <stdint.h>

typedef __attribute__((ext_vector_type(2))) float v2f;
typedef __attribute__((ext_vector_type(8))) float v8f;

#define IN_DIM  128
#define HID_DIM 128
#define OUT_DIM 64

// ---------------------------------------------------------------------------
// CDNA5 async global->LDS copy (ASYNCcnt path). LDS dest address is the low
// 32 bits of the generic pointer.
// ---------------------------------------------------------------------------
__device__ __forceinline__ void async_load_b32_to_lds(const float* lds_dst,
                                                      const float* gsrc) {
    uint32_t lds_off = (uint32_t)(uintptr_t)lds_dst;
    asm volatile("global_load_async_to_lds_b32 %0, %1, off"
                 :: "v"(lds_off), "v"(gsrc)
                 : "memory");
}

__device__ __forceinline__ void wait_async_zero() {
#if __has_builtin(__builtin_amdgcn_s_wait_asynccnt)
    __builtin_amdgcn_s_wait_asynccnt(0);
#else
    asm volatile("s_wait_asynccnt 0x0" ::: "memory");
#endif
}

// ---------------------------------------------------------------------------
// Zero workspace region (graph-capture safe; no hipMemset)
// ---------------------------------------------------------------------------
__global__ void gcn_zero_f32(float* __restrict__ p, int n) {
    int i = blockIdx.x * blockDim.x + threadIdx.x;
    if (i < n) p[i] = 0.0f;
}

// ---------------------------------------------------------------------------
// SpMM via scatter-add: one wave32 per edge, one float4 per lane (dim==128).
// agg[row] += val * h[col] with global_atomic_add_f32; the 51 MB agg buffer
// is L2-resident (192 MB), so the RMWs stay on-chip.
// ---------------------------------------------------------------------------
__global__ void gcn_spmm_atomic(const int*   __restrict__ rows,
                                const int*   __restrict__ cols,
                                const float* __restrict__ vals,
                                const float* __restrict__ h,
                                float*       __restrict__ agg,
                                int n_edges) {
    int wave = (blockIdx.x * blockDim.x + threadIdx.x) >> 5;
    int lane = threadIdx.x & 31;
    if (wave >= n_edges) return;

    int   r = rows[wave];
    int   c = cols[wave];
    float v = vals[wave];

    const float4* src = (const float4*)(h + (size_t)c * IN_DIM);
    float*        dst = agg + (size_t)r * IN_DIM + lane * 4;

    float4 m = src[lane];                    // 32 lanes * 16B = full 128-f32 row
    atomicAdd(dst + 0, v * m.x);
    atomicAdd(dst + 1, v * m.y);
    atomicAdd(dst + 2, v * m.z);
    atomicAdd(dst + 3, v * m.w);
}

// ---------------------------------------------------------------------------
// Dense GEMM  C[M,N] = A[M,K] * W[K,N]  (optionally ReLU) with
// V_WMMA_F32_16X16X4_F32.
//
// Block = 8 waves; the ENTIRE W (K*N*4 <= 64 KB, fits 320 KB/WGP LDS) is
// async-staged into LDS in a PAIR-INTERLEAVED layout:
//     element (k, n)  is stored at Bs[((k>>1)*N + n)*2 + (k&1)]
// so each B fragment {W[k][n], W[k+1][n]} is one aligned ds_load_b64
// (no cross-fragment mov fixups).  Each wave owns one 16-row A tile and
// computes ALL N output columns with N/16 accumulator fragments in VGPRs,
// so every A element is read from global exactly once.
//
// f32 WMMA fragment layouts (ISA 7.12.2):
//   A 16x4 :  lanes 0-15 row M=lane, K={0,1} in v0,v1 ; lanes 16-31 K={2,3}
//   B 4x16 :  v0: lanes0-15 K=0/N=lane, lanes16-31 K=2 ; v1: K=1 / K=3
//   C 16x16:  v_r: lanes0-15 M=r, lanes16-31 M=r+8 ; N = lane%16
// ---------------------------------------------------------------------------
template<bool RELU, int K, int N>
__global__ void gcn_gemm_wmma(const float* __restrict__ A,
                              const float* __restrict__ W,
                              float*       __restrict__ C,
                              int M) {
    constexpr int NT = N / 16;               // column tiles per wave
    __shared__ float Bs[K * N];              // pair-interleaved weight image

    // Async-stage W into LDS (contiguous global reads, permuted LDS writes),
    // drain ASYNCcnt, barrier.
    for (int i = threadIdx.x; i < K * N; i += blockDim.x) {
        int k = i / N;                       // N is a power of two -> shifts
        int n = i % N;
        int off = ((k >> 1) * N + n) * 2 + (k & 1);
        async_load_b32_to_lds(&Bs[off], &W[i]);
    }
    wait_async_zero();
    __syncthreads();

    const int waveId = threadIdx.x >> 5;
    const int lane   = threadIdx.x & 31;
    const int row0   = (blockIdx.x * 8 + waveId) * 16;
    if (row0 >= M) return;                    // wave-uniform: EXEC stays all-1s

    const int lrow  = lane & 15;              // M (for A) / N (for B) within tile
    const int khalf = (lane >> 4) << 1;       // 0 for lanes 0-15, 2 for 16-31

    v8f acc[NT];
#pragma unroll
    for (int n = 0; n < NT; ++n) acc[n] = (v8f){};

    const float* arow = A + (size_t)(row0 + lrow) * K + khalf;

#pragma unroll
    for (int k = 0; k < K; k += 4) {
        // A fragment loaded ONCE per k-step, reused for all NT column tiles.
        float2 af = *(const float2*)(arow + k);   // 8B aligned (khalf even)
        v2f a; a.x = af.x; a.y = af.y;

        const int kk = k + khalf;                 // even
        const float2* brow = (const float2*)&Bs[(kk >> 1) * N * 2];
#pragma unroll
        for (int n = 0; n < NT; ++n) {
            float2 bf = brow[n * 16 + lrow];      // one aligned ds_load_b64
            v2f b; b.x = bf.x; b.y = bf.y;
            acc[n] = __builtin_amdgcn_wmma_f32_16x16x4_f32(
                /*neg_a=*/false, a, /*neg_b=*/false, b,
                /*c_mod=*/(short)0, acc[n], /*reuse_a=*/false, /*reuse_b=*/false);
        }
    }

    const int cm = (lane >> 4) << 3;          // +8 rows for upper half-wave
    const int cn = lane & 15;
#pragma unroll
    for (int n = 0; n < NT; ++n) {
#pragma unroll
        for (int r = 0; r < 8; ++r) {
            float v = acc[n][r];
            if (RELU) v = fmaxf(v, 0.0f);
            C[(size_t)(row0 + cm + r) * N + n * 16 + cn] = v;
        }
    }
}

// ---------------------------------------------------------------------------
// Link decode: one wave32 per pair; lane holds float2 (dim==64); shuffle-reduce.
// ---------------------------------------------------------------------------
__global__ void gcn_decode(const int*   __restrict__ pairs,
                           const float* __restrict__ h,
                           float*       __restrict__ out,
                           int n_pairs) {
    int wave = (blockIdx.x * blockDim.x + threadIdx.x) >> 5;
    int lane = threadIdx.x & 31;
    if (wave >= n_pairs) return;

    int ia = pairs[2 * wave + 0];
    int ib = pairs[2 * wave + 1];

    const float2* pa = (const float2*)(h + (size_t)ia * OUT_DIM);
    const float2* pb = (const float2*)(h + (size_t)ib * OUT_DIM);
    float2 va = pa[lane];                     // 32 lanes * 2 f32 = 64 dims
    float2 vb = pb[lane];
    float s = va.x * vb.x + va.y * vb.y;

#pragma unroll
    for (int off = 16; off > 0; off >>= 1)
        s += __shfl_down(s, off, 32);

    if (lane == 0) out[wave] = s;
}

// ---------------------------------------------------------------------------
extern "C" void kernel_launch(void* const* d_in, const int* in_sizes, int n_in,
                              void* d_out, int out_size, void* d_ws, size_t ws_size,
                              hipStream_t stream) {
    const float* x     = (const float*)d_in[0];
    const int*   erow  = (const int*)  d_in[1];
    const int*   ecol  = (const int*)  d_in[2];
    const float* eval_ = (const float*)d_in[3];
    const int*   pairs = (const int*)  d_in[4];
    const float* W1    = (const float*)d_in[5];
    const float* W2    = (const float*)d_in[6];
    float*       out   = (float*)d_out;

    const int N = in_sizes[0] / IN_DIM;       // 100000
    const int E = in_sizes[1];                // 800000
    const int P = in_sizes[4] / 2;            // 500000

    // Workspace layout: [agg1 | agg2 | h1 | h2]  (agg1,agg2 adjacent -> one zero pass)
    float* agg1 = (float*)d_ws;
    float* agg2 = agg1 + (size_t)N * HID_DIM;
    float* h1   = agg2 + (size_t)N * HID_DIM;
    float* h2   = h1   + (size_t)N * HID_DIM;

    const int mtiles = (N + 15) / 16;         // 6250

    // 1) zero both aggregation buffers (one pass)
    {
        int n = N * HID_DIM * 2;
        gcn_zero_f32<<<(n + 255) / 256, 256, 0, stream>>>(agg1, n);
    }
    // 2) SpMM layer 1: agg1 = scatter(vals * x[cols])
    gcn_spmm_atomic<<<(E * 32 + 255) / 256, 256, 0, stream>>>(erow, ecol, eval_, x, agg1, E);
    // 3) h1 = relu(agg1 @ W1)   [K=128, N=128]
    gcn_gemm_wmma<true, IN_DIM, HID_DIM>
        <<<(mtiles + 7) / 8, 256, 0, stream>>>(agg1, W1, h1, N);
    // 4) SpMM layer 2: agg2 = scatter(vals * h1[cols])
    gcn_spmm_atomic<<<(E * 32 + 255) / 256, 256, 0, stream>>>(erow, ecol, eval_, h1, agg2, E);
    // 5) h2 = agg2 @ W2   [K=128, N=64]
    gcn_gemm_wmma<false, HID_DIM, OUT_DIM>
        <<<(mtiles + 7) / 8, 256, 0, stream>>>(agg2, W2, h2, N);
    // 6) decode pairs
    gcn_decode<<<(P * 32 + 255) / 256, 256, 0, stream>>>(pairs, h2, out, P);
}